// GCNEncoder_17463337025661
// MI455X (gfx1250) — compile-verified
//
#include <hip/hip_runtime.h>
#include <cstdint>
#include <cstddef>

typedef float v2f __attribute__((ext_vector_type(2)));
typedef float v8f __attribute__((ext_vector_type(8)));

// ---------------------------------------------------------------------------
// fp32 atomic add, relaxed, device scope -> global_atomic_add_f32
// ---------------------------------------------------------------------------
__device__ __forceinline__ void atomicAddF32(float* p, float v) {
    __hip_atomic_fetch_add(p, v, __ATOMIC_RELAXED, __HIP_MEMORY_SCOPE_AGENT);
}

// ---------------------------------------------------------------------------
// Zero-fill (float4 granularity; all sizes here are multiples of 4 floats)
// ---------------------------------------------------------------------------
__global__ void zero_kernel(float4* __restrict__ p, long long n4) {
    long long i = (long long)blockIdx.x * blockDim.x + threadIdx.x;
    if (i < n4) p[i] = make_float4(0.f, 0.f, 0.f, 0.f);
}

// ---------------------------------------------------------------------------
// Degree / normalization: deg starts at 1.0 (self loop), += 1 per dst edge,
// then deg <- rsqrt(deg)
// ---------------------------------------------------------------------------
__global__ void deg_init_kernel(float* __restrict__ deg, int n) {
    int i = blockIdx.x * blockDim.x + threadIdx.x;
    if (i < n) deg[i] = 1.0f;
}

__global__ void deg_accum_kernel(const int* __restrict__ dst, float* __restrict__ deg, int e) {
    int i = blockIdx.x * blockDim.x + threadIdx.x;
    if (i < e) atomicAddF32(&deg[dst[i]], 1.0f);
}

__global__ void dinv_kernel(float* __restrict__ deg, int n) {
    int i = blockIdx.x * blockDim.x + threadIdx.x;
    if (i < n) {
        float d = deg[i];
        deg[i] = (d > 0.f) ? rsqrtf(d) : 0.f;
    }
}

// ---------------------------------------------------------------------------
// H[N,Co] = X[N,K] @ W[K,Co] using V_WMMA_F32_16X16X4_F32.
// One wave owns a 16-row x 64-col tile: 4 v8f accumulators.
// Fragment layouts per CDNA5 ISA 7.12.2 (32-bit A 16x4, 32-bit C/D 16x16):
//   A: lanes 0-15 hold row M=lane, K = k..k+1; lanes 16-31 hold K = k+2..k+3
//   B: VGPR j, lanes 0-15 -> row K=k+2*half (cols 0-15), mirrored like A
//   C: VGPR v, lanes 0-15 -> M=v, N=lane; lanes 16-31 -> M=v+8, N=lane-16
// All addressing uses 32-bit offsets off a uniform base pointer so loads
// lower to saddr-form global_load with a 32-bit voffset (no v_add_nc_u64).
// Requires N % 16 == 0, K % 4 == 0, Co % 64 == 0 (all hold for this problem).
// ---------------------------------------------------------------------------
__global__ void gemm_wmma_f32(const float* __restrict__ X,
                              const float* __restrict__ W,
                              float* __restrict__ H,
                              int Nrows, int K, int Co) {
    const int lane         = threadIdx.x & 31;
    const int waveInBlock  = threadIdx.x >> 5;
    const int wavesPerBlk  = blockDim.x >> 5;
    const int wave         = blockIdx.x * wavesPerBlk + waveInBlock;

    const int colGroups = Co >> 6;       // 64 output cols per wave
    const int rowTiles  = Nrows >> 4;    // 16 rows per tile
    if (wave >= rowTiles * colGroups) return;   // wave-uniform -> EXEC all 1s below

    const int rowTile  = wave / colGroups;
    const int colGroup = wave - rowTile * colGroups;

    const int half    = lane >> 4;       // 0: K=k..k+1, 1: K=k+2..k+3
    const int l15     = lane & 15;
    const int aRow    = rowTile * 16 + l15;
    const int colBase = colGroup * 64;

    v8f acc0 = {}, acc1 = {}, acc2 = {}, acc3 = {};

    // 32-bit running offsets (max X offset ~ 25.6M floats -> fits easily)
    int xOff = aRow * K + 2 * half;              // advances by 4 per k-step
    int wOff = (2 * half) * Co + colBase + l15;  // advances by 4*Co per k-step
    const int wStep = 4 * Co;

    for (int k = 0; k < K; k += 4) {
        // A fragment: two consecutive K values of this lane's row (8B aligned)
        v2f a = *(const v2f*)(X + xOff);
        // B fragments for the 4 column sub-tiles (coalesced across lanes 0-15)
        v2f b0, b1, b2, b3;
        b0[0] = W[wOff];      b0[1] = W[wOff + Co];
        b1[0] = W[wOff + 16]; b1[1] = W[wOff + Co + 16];
        b2[0] = W[wOff + 32]; b2[1] = W[wOff + Co + 32];
        b3[0] = W[wOff + 48]; b3[1] = W[wOff + Co + 48];

        acc0 = __builtin_amdgcn_wmma_f32_16x16x4_f32(false, a, false, b0, (short)0, acc0, false, false);
        acc1 = __builtin_amdgcn_wmma_f32_16x16x4_f32(false, a, false, b1, (short)0, acc1, false, false);
        acc2 = __builtin_amdgcn_wmma_f32_16x16x4_f32(false, a, false, b2, (short)0, acc2, false, false);
        acc3 = __builtin_amdgcn_wmma_f32_16x16x4_f32(false, a, false, b3, (short)0, acc3, false, false);

        xOff += 4;
        wOff += wStep;
    }

    // Write back per C/D layout
    int cOff = (rowTile * 16 + 8 * half) * Co + colBase + l15;
    #pragma unroll
    for (int v = 0; v < 8; ++v) {
        H[cOff]      = acc0[v];
        H[cOff + 16] = acc1[v];
        H[cOff + 32] = acc2[v];
        H[cOff + 48] = acc3[v];
        cOff += Co;
    }
}

// ---------------------------------------------------------------------------
// Edge scatter: agg[dst] += h[src] * dinv[src]*dinv[dst]
// C/4 consecutive threads per edge -> coalesced float4 gather + 4 atomics.
// C is a compile-time constant so the edge/chunk split is pure shifts.
// h and agg both fit in the 192MB L2, so these atomics resolve in-L2.
// ---------------------------------------------------------------------------
template <int C>
__global__ void scatter_edges(const float* __restrict__ h,
                              const float* __restrict__ dinv,
                              const int*   __restrict__ src,
                              const int*   __restrict__ dst,
                              float* __restrict__ agg,
                              int e) {
    constexpr int CHUNKS = C >> 2;             // float4 chunks per row (64 or 32)
    long long tid   = (long long)blockIdx.x * blockDim.x + threadIdx.x;
    long long total = (long long)e * CHUNKS;
    if (tid >= total) return;
    int edge  = (int)(tid / CHUNKS);           // constant power of two -> shift
    int chunk = (int)(tid & (CHUNKS - 1));

    int s = src[edge];
    int d = dst[edge];
    float nrm = dinv[s] * dinv[d];

    float4 hv = *(const float4*)(h + (size_t)s * C + chunk * 4);
    float* out = agg + (size_t)d * C + chunk * 4;
    atomicAddF32(out + 0, hv.x * nrm);
    atomicAddF32(out + 1, hv.y * nrm);
    atomicAddF32(out + 2, hv.z * nrm);
    atomicAddF32(out + 3, hv.w * nrm);
}

// ---------------------------------------------------------------------------
// Finalize (in place): agg[i] = act(agg[i] + dinv[i]^2 * h[i] + b)
// Folds the self-loop edge (norm = dinv[i]*dinv[i]) so we never materialize it.
// ---------------------------------------------------------------------------
template <int C, bool RELU>
__global__ void finalize_kernel(float* __restrict__ agg,
                                const float* __restrict__ h,
                                const float* __restrict__ dinv,
                                const float* __restrict__ bias,
                                int n) {
    constexpr int CHUNKS = C >> 2;
    long long tid   = (long long)blockIdx.x * blockDim.x + threadIdx.x;
    long long total = (long long)n * CHUNKS;
    if (tid >= total) return;
    int i     = (int)(tid / CHUNKS);
    int chunk = (int)(tid & (CHUNKS - 1));

    float di = dinv[i];
    float w  = di * di;

    size_t o = (size_t)i * C + chunk * 4;
    float4 a  = *(const float4*)(agg + o);
    float4 hv = *(const float4*)(h + o);
    float4 bv = *(const float4*)(bias + chunk * 4);

    float4 r;
    r.x = fmaf(w, hv.x, a.x) + bv.x;
    r.y = fmaf(w, hv.y, a.y) + bv.y;
    r.z = fmaf(w, hv.z, a.z) + bv.z;
    r.w = fmaf(w, hv.w, a.w) + bv.w;
    if (RELU) {
        r.x = fmaxf(r.x, 0.f); r.y = fmaxf(r.y, 0.f);
        r.z = fmaxf(r.z, 0.f); r.w = fmaxf(r.w, 0.f);
    }
    *(float4*)(agg + o) = r;
}

// ---------------------------------------------------------------------------
// Host-side orchestration
// ---------------------------------------------------------------------------
static inline unsigned cdiv_u(long long a, long long b) { return (unsigned)((a + b - 1) / b); }

extern "C" void kernel_launch(void* const* d_in, const int* in_sizes, int n_in,
                              void* d_out, int out_size, void* d_ws, size_t ws_size,
                              hipStream_t stream) {
    const float* x  = (const float*)d_in[0];
    const int*   ei = (const int*)d_in[1];
    const float* W0 = (const float*)d_in[2];
    const float* b0 = (const float*)d_in[3];
    const float* W1 = (const float*)d_in[4];
    const float* b1 = (const float*)d_in[5];
    const float* W2 = (const float*)d_in[6];
    const float* b2 = (const float*)d_in[7];

    const int N = in_sizes[0] / 512;   // 50000
    const int E = in_sizes[1] / 2;     // 1,600,000
    const int* src = ei;
    const int* dst = ei + E;

    // Workspace layout: dinv | bufA (N*256) | bufB (N*256)
    char* ws = (char*)d_ws;
    size_t off = 0;
    float* dinv = (float*)(ws + off); off += (((size_t)N * 4) + 255) & ~(size_t)255;
    float* bufA = (float*)(ws + off); off += (size_t)N * 256 * 4;
    float* bufB = (float*)(ws + off);

    const int TB = 256;
    float* outp = (float*)d_out;

    // --- normalization -----------------------------------------------------
    deg_init_kernel<<<cdiv_u(N, TB), TB, 0, stream>>>(dinv, N);
    deg_accum_kernel<<<cdiv_u(E, TB), TB, 0, stream>>>(dst, dinv, E);
    dinv_kernel<<<cdiv_u(N, TB), TB, 0, stream>>>(dinv, N);

    // --- layer 0: 512 -> 256, relu ----------------------------------------
    {
        const int K = 512, Co = 256;
        unsigned waves = (unsigned)((N >> 4) * (Co >> 6));
        gemm_wmma_f32<<<cdiv_u(waves, TB / 32), TB, 0, stream>>>(x, W0, bufA, N, K, Co);
        zero_kernel<<<cdiv_u((long long)N * Co / 4, TB), TB, 0, stream>>>((float4*)bufB, (long long)N * Co / 4);
        scatter_edges<256><<<cdiv_u((long long)E * (Co / 4), TB), TB, 0, stream>>>(bufA, dinv, src, dst, bufB, E);
        finalize_kernel<256, true><<<cdiv_u((long long)N * Co / 4, TB), TB, 0, stream>>>(bufB, bufA, dinv, b0, N);
    }

    // --- layer 1: 256 -> 256, relu ----------------------------------------
    {
        const int K = 256, Co = 256;
        unsigned waves = (unsigned)((N >> 4) * (Co >> 6));
        gemm_wmma_f32<<<cdiv_u(waves, TB / 32), TB, 0, stream>>>(bufB, W1, bufA, N, K, Co);
        zero_kernel<<<cdiv_u((long long)N * Co / 4, TB), TB, 0, stream>>>((float4*)bufB, (long long)N * Co / 4);
        scatter_edges<256><<<cdiv_u((long long)E * (Co / 4), TB), TB, 0, stream>>>(bufA, dinv, src, dst, bufB, E);
        finalize_kernel<256, true><<<cdiv_u((long long)N * Co / 4, TB), TB, 0, stream>>>(bufB, bufA, dinv, b1, N);
    }

    // --- layer 2: 256 -> 128, linear, scatter straight into d_out ---------
    {
        const int K = 256, Co = 128;
        unsigned waves = (unsigned)((N >> 4) * (Co >> 6));
        gemm_wmma_f32<<<cdiv_u(waves, TB / 32), TB, 0, stream>>>(bufB, W2, bufA, N, K, Co);
        zero_kernel<<<cdiv_u((long long)N * Co / 4, TB), TB, 0, stream>>>((float4*)outp, (long long)N * Co / 4);
        scatter_edges<128><<<cdiv_u((long long)E * (Co / 4), TB), TB, 0, stream>>>(bufA, dinv, src, dst, outp, E);
        finalize_kernel<128, false><<<cdiv_u((long long)N * Co / 4, TB), TB, 0, stream>>>(outp, bufA, dinv, b2, N);
    }
}